// GATLayer_37056977830238
// MI455X (gfx1250) — compile-verified
//
#include <hip/hip_runtime.h>
#include <hip/hip_bf16.h>

typedef __attribute__((ext_vector_type(2))) float v2f;
typedef __attribute__((ext_vector_type(8))) float v8f;
typedef __attribute__((ext_vector_type(2))) int   v2i;

#define NROWS   8192
#define INF_    256
#define OUTF    64
#define ALPHA   0.2f
#define NSPLIT  4                       // j-range splits per row tile
#define JSLAB   (NROWS / NSPLIT)        // 2048
#define PSTRIDE 1040                    // 16*64 acc + 16 denom per partial

#define AS_GLOBAL __attribute__((address_space(1)))
#define AS_LOCAL  __attribute__((address_space(3)))

// D = A(16x4 f32) * B(4x16 f32) + C(16x16 f32), wave32, exact f32
__device__ __forceinline__ v8f wmma_f32(v2f a, v2f b, v8f c) {
    return __builtin_amdgcn_wmma_f32_16x16x4_f32(false, a, false, b, (short)0, c,
                                                 false, false);
}

__device__ __forceinline__ float elu_f(float x) {
    return x > 0.0f ? x : expm1f(x);
}

// ---------------------------------------------------------------------------
// Phase 1: Ht = (X @ W)^T stored column-major (64 x 8192).
// One wave per 16-row tile; C-layout rows are consecutive VGPRs per lane,
// so the transposed store vectorizes to b128.
// ---------------------------------------------------------------------------
__global__ __launch_bounds__(256) void gat_h_gemm(const float* __restrict__ X,
                                                  const float* __restrict__ W,
                                                  float* __restrict__ Ht) {
    const int lane  = threadIdx.x & 31;
    const int wave  = threadIdx.x >> 5;
    const int tile  = blockIdx.x * 8 + wave;      // 512 tiles
    const int row0  = tile * 16;
    const int mrow  = lane & 15;
    const int khalf = (lane >> 4) * 2;            // K sub-slot (0 or 2)

    v8f acc0 = {}, acc1 = {}, acc2 = {}, acc3 = {};
    const float* xrow = X + (size_t)(row0 + mrow) * INF_;

#pragma unroll 4
    for (int k = 0; k < INF_; k += 4) {
        const float2 av = *(const float2*)(xrow + k + khalf);   // K, K+1
        v2f a; a.x = av.x; a.y = av.y;

        const float* w0 = W + (size_t)(k + khalf) * OUTF + mrow;
        const float* w1 = w0 + OUTF;
        v2f b0; b0.x = w0[0];  b0.y = w1[0];
        v2f b1; b1.x = w0[16]; b1.y = w1[16];
        v2f b2; b2.x = w0[32]; b2.y = w1[32];
        v2f b3; b3.x = w0[48]; b3.y = w1[48];

        acc0 = wmma_f32(a, b0, acc0);
        acc1 = wmma_f32(a, b1, acc1);
        acc2 = wmma_f32(a, b2, acc2);
        acc3 = wmma_f32(a, b3, acc3);
    }

    // Transposed store: Ht[col * 8192 + row], rows row0+cr .. row0+cr+7
    const int cr = (lane >> 4) * 8;
    const int cc = lane & 15;
    const size_t rbase = (size_t)row0 + cr;
    float4 lo, hi;
    lo = make_float4(acc0[0], acc0[1], acc0[2], acc0[3]);
    hi = make_float4(acc0[4], acc0[5], acc0[6], acc0[7]);
    *(float4*)(Ht + (size_t)(cc +  0) * NROWS + rbase)     = lo;
    *(float4*)(Ht + (size_t)(cc +  0) * NROWS + rbase + 4) = hi;
    lo = make_float4(acc1[0], acc1[1], acc1[2], acc1[3]);
    hi = make_float4(acc1[4], acc1[5], acc1[6], acc1[7]);
    *(float4*)(Ht + (size_t)(cc + 16) * NROWS + rbase)     = lo;
    *(float4*)(Ht + (size_t)(cc + 16) * NROWS + rbase + 4) = hi;
    lo = make_float4(acc2[0], acc2[1], acc2[2], acc2[3]);
    hi = make_float4(acc2[4], acc2[5], acc2[6], acc2[7]);
    *(float4*)(Ht + (size_t)(cc + 32) * NROWS + rbase)     = lo;
    *(float4*)(Ht + (size_t)(cc + 32) * NROWS + rbase + 4) = hi;
    lo = make_float4(acc3[0], acc3[1], acc3[2], acc3[3]);
    hi = make_float4(acc3[4], acc3[5], acc3[6], acc3[7]);
    *(float4*)(Ht + (size_t)(cc + 48) * NROWS + rbase)     = lo;
    *(float4*)(Ht + (size_t)(cc + 48) * NROWS + rbase + 4) = hi;
}

// ---------------------------------------------------------------------------
// Phase 2: f1 = H @ a[:64], f2 = H @ a[64:]  — reads Ht fully coalesced.
// ---------------------------------------------------------------------------
__global__ __launch_bounds__(256) void gat_f(const float* __restrict__ Ht,
                                             const float* __restrict__ a,
                                             float* __restrict__ f1,
                                             float* __restrict__ f2) {
    const int i = blockIdx.x * blockDim.x + threadIdx.x;
    float s1 = 0.0f, s2 = 0.0f;
#pragma unroll 8
    for (int k = 0; k < OUTF; ++k) {
        const float v = Ht[(size_t)k * NROWS + i];
        s1 = fmaf(v, a[k], s1);
        s2 = fmaf(v, a[OUTF + k], s2);
    }
    f1[i] = s1;
    f2[i] = s2;
}

// ---------------------------------------------------------------------------
// Phase 3: scalar max(f2) (single block) — safe global softmax shift, since
// leakyrelu is monotone: max_j e_ij = leakyrelu(f1_i + max_j f2_j).
// ---------------------------------------------------------------------------
__global__ __launch_bounds__(1024) void gat_maxf2(const float* __restrict__ f2,
                                                  float* __restrict__ mx) {
    __shared__ float red[1024];
    float m = -__builtin_inff();
    for (int i = threadIdx.x; i < NROWS; i += 1024) m = fmaxf(m, f2[i]);
    red[threadIdx.x] = m;
    __syncthreads();
    for (int s = 512; s > 0; s >>= 1) {
        if (threadIdx.x < s)
            red[threadIdx.x] = fmaxf(red[threadIdx.x], red[threadIdx.x + s]);
        __syncthreads();
    }
    if (threadIdx.x == 0) *mx = red[0];
}

// ---------------------------------------------------------------------------
// Phase 4: fused masked softmax + partial (attn @ H).
// 2048 waves: 512 row tiles x 4 j-slabs (all slabs share the same shift m_i,
// so partials combine additively). 5 f32 WMMAs per 4-j step: 4 output tiles
// + one all-ones B accumulating the denominator in matching C layout.
// adj is a consume-once 256MB stream -> non-temporal loads.
// f2 staged to LDS via async global->LDS copies (ASYNCcnt) when available.
// ---------------------------------------------------------------------------
__global__ __launch_bounds__(256) void gat_attn(const int*   __restrict__ adj,
                                                const float* __restrict__ Ht,
                                                const float* __restrict__ f1,
                                                const float* __restrict__ f2,
                                                const float* __restrict__ mxp,
                                                float* __restrict__ part) {
    __shared__ float sf2[NROWS];
#if defined(__AMDGCN__) && __has_builtin(__builtin_amdgcn_global_load_async_to_lds_b32)
    for (int i = threadIdx.x; i < NROWS; i += 256) {
        __builtin_amdgcn_global_load_async_to_lds_b32(
            (AS_GLOBAL int*)(f2 + i), (AS_LOCAL int*)(sf2 + i), 0, 0);
    }
#if __has_builtin(__builtin_amdgcn_s_wait_asynccnt)
    __builtin_amdgcn_s_wait_asynccnt(0);
#else
    asm volatile("s_wait_asynccnt 0x0" ::: "memory");
#endif
#else
    for (int i = threadIdx.x; i < NROWS; i += 256) sf2[i] = f2[i];
#endif
    __syncthreads();

    const int lane  = threadIdx.x & 31;
    const int wave  = threadIdx.x >> 5;
    const int gwave = blockIdx.x * 8 + wave;      // 0..2047
    const int tile  = gwave >> 2;                 // 0..511
    const int q     = gwave & 3;                  // j-slab
    const int row0  = tile * 16;
    const int j0    = q * JSLAB;
    const int mrow  = lane & 15;
    const int khalf = (lane >> 4) * 2;

    const float fi = f1[row0 + mrow];
    const float mx = *mxp;
    const float ms = fi + mx;
    const float mi = fmaxf(ms, ALPHA * ms);       // >= masked row max

    v8f acc0 = {}, acc1 = {}, acc2 = {}, acc3 = {}, accl = {};
    const v2f ones = {1.0f, 1.0f};
    const int*   arow = adj + (size_t)(row0 + mrow) * NROWS + j0;
    const float* hc   = Ht + mrow;                // + n0*8192 selects N-tile col

#pragma unroll 2
    for (int j = 0; j < JSLAB; j += 4) {
        const int jj = j + khalf;                 // slab-local K pair base

        const v2i    av = __builtin_nontemporal_load((const v2i*)(arow + jj));
        const float2 fv = *(const float2*)(sf2 + j0 + jj);    // ds_load_b64
        __builtin_prefetch(arow + jj + 512, 0, 0);            // global_prefetch

        const float e0x = fi + fv.x;
        const float e1x = fi + fv.y;
        const float e0  = fmaxf(e0x, ALPHA * e0x);            // leakyrelu
        const float e1  = fmaxf(e1x, ALPHA * e1x);
        v2f pa;
        pa.x = (av.x > 0) ? __expf(e0 - mi) : 0.0f;
        pa.y = (av.y > 0) ? __expf(e1 - mi) : 0.0f;

        const size_t jg = (size_t)(j0 + jj);                  // global j
        const float2 h0 = *(const float2*)(hc + (size_t) 0 * NROWS + jg);
        const float2 h1 = *(const float2*)(hc + (size_t)16 * NROWS + jg);
        const float2 h2 = *(const float2*)(hc + (size_t)32 * NROWS + jg);
        const float2 h3 = *(const float2*)(hc + (size_t)48 * NROWS + jg);
        v2f b0; b0.x = h0.x; b0.y = h0.y;
        v2f b1; b1.x = h1.x; b1.y = h1.y;
        v2f b2; b2.x = h2.x; b2.y = h2.y;
        v2f b3; b3.x = h3.x; b3.y = h3.y;

        acc0 = wmma_f32(pa, b0, acc0);
        acc1 = wmma_f32(pa, b1, acc1);
        acc2 = wmma_f32(pa, b2, acc2);
        acc3 = wmma_f32(pa, b3, acc3);
        accl = wmma_f32(pa, ones, accl);                      // denominator
    }

    // Write partials: acc(16x64) then denom(16)
    float* p = part + ((size_t)tile * NSPLIT + q) * PSTRIDE;
    const int cr = (lane >> 4) * 8;
    const int cc = lane & 15;
#pragma unroll
    for (int r = 0; r < 8; ++r) {
        float* o = p + (size_t)(cr + r) * OUTF + cc;
        o[0]  = acc0[r];
        o[16] = acc1[r];
        o[32] = acc2[r];
        o[48] = acc3[r];
    }
    if (cc == 0) {
#pragma unroll
        for (int r = 0; r < 8; ++r) p[1024 + cr + r] = accl[r];
    }
}

// ---------------------------------------------------------------------------
// Phase 5: combine 4 j-slab partials, normalize, elu.
// ---------------------------------------------------------------------------
__global__ __launch_bounds__(256) void gat_combine(const float* __restrict__ part,
                                                   float* __restrict__ out) {
    const int idx  = blockIdx.x * 256 + threadIdx.x;   // 524288
    const int tile = idx >> 10;
    const int elem = idx & 1023;
    const int row  = elem >> 6;
    const float* p = part + (size_t)tile * NSPLIT * PSTRIDE;
    float s = 0.0f, l = 0.0f;
#pragma unroll
    for (int q = 0; q < NSPLIT; ++q) {
        s += p[q * PSTRIDE + elem];
        l += p[q * PSTRIDE + 1024 + row];
    }
    out[(size_t)tile * 1024 + elem] = elu_f(s / l);
}

// ---------------------------------------------------------------------------
extern "C" void kernel_launch(void* const* d_in, const int* in_sizes, int n_in,
                              void* d_out, int out_size, void* d_ws, size_t ws_size,
                              hipStream_t stream) {
    const float* input_h = (const float*)d_in[0];   // (8192, 256)
    const int*   adj     = (const int*)d_in[1];     // (8192, 8192)
    const float* W       = (const float*)d_in[2];   // (256, 64)
    const float* a       = (const float*)d_in[3];   // (128, 1)
    float*       out     = (float*)d_out;           // (8192, 64)

    float* Ht   = (float*)d_ws;                     // 64*8192 (transposed H)
    float* f1   = Ht + (size_t)OUTF * NROWS;        // 8192
    float* f2   = f1 + NROWS;                       // 8192
    float* mx   = f2 + NROWS;                       // pad to 8
    float* part = mx + 8;                           // 512*4*1040

    gat_h_gemm<<<64, 256, 0, stream>>>(input_h, W, Ht);
    gat_f<<<NROWS / 256, 256, 0, stream>>>(Ht, a, f1, f2);
    gat_maxf2<<<1, 1024, 0, stream>>>(f2, mx);
    gat_attn<<<256, 256, 0, stream>>>(adj, Ht, f1, f2, mx, part);
    gat_combine<<<2048, 256, 0, stream>>>(part, out);
}